// FastRCNNOutputLayers_36335423324754
// MI455X (gfx1250) — compile-verified
//
#include <hip/hip_runtime.h>
#include <hip/hip_bf16.h>
#include <math.h>

typedef float v2f __attribute__((ext_vector_type(2)));
typedef float v4f __attribute__((ext_vector_type(4)));
typedef float v8f __attribute__((ext_vector_type(8)));
typedef unsigned int v4u __attribute__((ext_vector_type(4)));
typedef int v4i __attribute__((ext_vector_type(4)));
typedef int v8i __attribute__((ext_vector_type(8)));

#define B_IMG 16
#define N_PROP 1024
#define D_IN 1024
#define K_CLS 80
#define M_ROWS (B_IMG * N_PROP)          // 16384
#define LOGITS_LD 96                     // 81 padded to 6 wmma col-tiles
#define DELTAS_LD 320
#define M_PRE 1024
#define TOPK 100
#define CAND_PER_IMG (N_PROP * K_CLS)    // 81920
#define IMG_Wc 1333.0f
#define IMG_Hc 800.0f
#define SCORE_THRESH 0.05f
#define NMS_THRESH 0.5f
#define SCALE_CLAMP 4.135166556742356f

// ---------------------------------------------------------------------------
// TDM helper: issue a 2D tile load (fp32 elements) global -> LDS with
// hardware row padding: rows of `tile0` DWORDs padded by 4 DWORDs in LDS
// (pad_interval=5 -> 64 DWORDs, pad_amount=3 -> 4 DWORDs => LDS stride 68).
// tensor_dim* are the remaining extents from the tile origin; rows/cols past
// them are zero-filled by TDM OOB handling (used for the padded cls tile).
// ---------------------------------------------------------------------------
__device__ __forceinline__ void tdm_load_2d(
    unsigned lds_off, unsigned long long gaddr,
    unsigned dim0_rem, unsigned dim1_rem,
    unsigned tile0, unsigned tile1, unsigned stride0)
{
    v4u g0;
    g0.x = 1u;                                                  // count=1 (valid)
    g0.y = lds_off;                                             // LDS byte address
    g0.z = (unsigned)(gaddr & 0xFFFFFFFFull);                   // global addr lo
    g0.w = (unsigned)((gaddr >> 32) & 0x1FFFFFFull) | (2u << 30); // hi | type=2
    v8i g1;
    g1[0] = (int)((2u << 16)      // data_size = 4 bytes
                | (1u << 20)      // pad_enable
                | (5u << 22)      // pad_interval: 64 DWORDs
                | (3u << 25));    // pad_amount: 4 DWORDs
    g1[1] = (int)((dim0_rem & 0xFFFFu) << 16);                  // tensor_dim0[15:0]
    g1[2] = (int)((dim0_rem >> 16) | ((dim1_rem & 0xFFFFu) << 16));
    g1[3] = (int)((dim1_rem >> 16) | (tile0 << 16));            // tile_dim0
    g1[4] = (int)tile1;                                         // tile_dim1, tile_dim2=0
    g1[5] = (int)stride0;                                       // dim0 stride lo
    g1[6] = 0;                                                  // stride hi / dim1 stride
    g1[7] = 0;
    v4i gz4 = {0, 0, 0, 0};
    v8i gz8 = {0, 0, 0, 0, 0, 0, 0, 0};
    __builtin_amdgcn_tensor_load_to_lds(g0, g1, gz4, gz4, gz8, 0);
}

// ---------------------------------------------------------------------------
// Kernel 1: fp32 GEMM  C[M, n_cols] = X[M,1024] * W[n_cols,1024]^T + bias
// 64x32 block tile, 8 wave32 waves, each wave one 16x16 tile via
// V_WMMA_F32_16X16X4_F32. Tiles staged by the Tensor Data Mover into
// double-buffered LDS (TENSORcnt-synchronized), stride-68 bank padding.
// ---------------------------------------------------------------------------
#define BM 64
#define BN 32
#define KC 64
#define LDSS 68

__global__ __launch_bounds__(256) void gemm_head(
    const float* __restrict__ X, const float* __restrict__ W,
    const float* __restrict__ bias, float* __restrict__ C,
    int n_cols, int ldc)
{
    __shared__ float Xs[2 * BM * LDSS];
    __shared__ float Ws[2 * BN * LDSS];
    const int tid   = threadIdx.x;
    const int lane  = tid & 31;
    const int wv    = tid >> 5;
    const int mBlk  = blockIdx.x * BM;
    const int nBase = blockIdx.y * BN;
    const int m0w   = (wv & 3) * 16;   // 4 waves along M
    const int n0w   = (wv >> 2) * 16;  // 2 waves along N

    const unsigned ldsX = (unsigned)(uintptr_t)&Xs[0];
    const unsigned ldsW = (unsigned)(uintptr_t)&Ws[0];
    const unsigned long long gX =
        (unsigned long long)(uintptr_t)X + (unsigned long long)mBlk * D_IN * 4ull;
    const unsigned long long gW =
        (unsigned long long)(uintptr_t)W + (unsigned long long)nBase * D_IN * 4ull;
    const unsigned wRows = (unsigned)(n_cols - nBase);  // <BN only for padded cls tile

    // prologue: DMA chunk 0 into buffer 0 (wave0: X tile, wave1: W tile)
    if (wv == 0)
        tdm_load_2d(ldsX, gX, D_IN, BM, KC, BM, D_IN);
    else if (wv == 1)
        tdm_load_2d(ldsW, gW, D_IN, wRows, KC, BN, D_IN);

    v8f acc = {0.f, 0.f, 0.f, 0.f, 0.f, 0.f, 0.f, 0.f};

    const int ar = lane & 15;          // A/B: row/col = lane%16
    const int ak = (lane >> 4) << 1;   // A/B: K pair  = 2*(lane/16)

    int ib = 0;
    for (int kc = 0; kc < D_IN; kc += KC, ib ^= 1) {
        // issue next chunk into the other buffer (overlaps with compute below)
        if (kc + KC < D_IN) {
            unsigned nb = (unsigned)(ib ^ 1);
            if (wv == 0)
                tdm_load_2d(ldsX + nb * (BM * LDSS * 4u),
                            gX + (unsigned long long)(kc + KC) * 4ull,
                            (unsigned)(D_IN - (kc + KC)), BM, KC, BM, D_IN);
            else if (wv == 1)
                tdm_load_2d(ldsW + nb * (BN * LDSS * 4u),
                            gW + (unsigned long long)(kc + KC) * 4ull,
                            (unsigned)(D_IN - (kc + KC)), wRows, KC, BN, D_IN);
            if (wv < 2) __builtin_amdgcn_s_wait_tensorcnt(1);  // oldest (current) done
        } else {
            if (wv < 2) __builtin_amdgcn_s_wait_tensorcnt(0);
        }
        __syncthreads();               // data visible to all waves

        const float* xb = Xs + ib * (BM * LDSS);
        const float* wb = Ws + ib * (BN * LDSS);
        #pragma unroll
        for (int kk = 0; kk < KC; kk += 4) {
            v2f a = *reinterpret_cast<const v2f*>(&xb[(m0w + ar) * LDSS + kk + ak]);
            v2f b = *reinterpret_cast<const v2f*>(&wb[(n0w + ar) * LDSS + kk + ak]);
            acc = __builtin_amdgcn_wmma_f32_16x16x4_f32(
                false, a, false, b, (short)0, acc, false, false);
        }
        __syncthreads();               // done reading before this buffer is refilled
    }

    // C/D layout: lane n = lane%16, row = 8*(lane/16) + vgpr
    int nCol = nBase + n0w + (lane & 15);
    int bcol = (nCol < n_cols) ? nCol : n_cols - 1;
    float bs = bias[bcol];
    int rowBase = mBlk + m0w + ((lane >> 4) << 3);
    #pragma unroll
    for (int r = 0; r < 8; ++r)
        C[(size_t)(rowBase + r) * ldc + nCol] = acc[r] + bs;
}

// ---------------------------------------------------------------------------
// Kernel 2: softmax over 81 logits (1 wave per row), drop bg, threshold ->
// order-preserving u32 sort keys; decode + clip boxes.
// ---------------------------------------------------------------------------
__global__ __launch_bounds__(256) void softmax_decode(
    const float* __restrict__ logits, const float* __restrict__ deltas,
    const float* __restrict__ proposals, float* __restrict__ boxes,
    unsigned int* __restrict__ cand)
{
    int wv   = threadIdx.x >> 5;
    int lane = threadIdx.x & 31;
    int row  = blockIdx.x * 8 + wv;        // [0, 16384)
    int b    = row >> 10;
    int n    = row & 1023;

    const float* lg = logits + (size_t)row * LOGITS_LD;
    float v0 = lg[lane];
    float v1 = lg[lane + 32];
    float v2 = (lane < 17) ? lg[lane + 64] : -3.4e38f;
    float mx = fmaxf(fmaxf(v0, v1), v2);
    #pragma unroll
    for (int off = 16; off; off >>= 1) mx = fmaxf(mx, __shfl_xor(mx, off, 32));
    float e0 = __expf(v0 - mx);
    float e1 = __expf(v1 - mx);
    float e2 = (lane < 17) ? __expf(v2 - mx) : 0.0f;
    float sm = e0 + e1 + e2;
    #pragma unroll
    for (int off = 16; off; off >>= 1) sm += __shfl_xor(sm, off, 32);
    float inv = 1.0f / sm;

    unsigned int* cr = cand + (size_t)b * CAND_PER_IMG + (size_t)n * K_CLS;
    auto emit = [&](int c, float e) {
        float p = e * inv;
        float s = (p > SCORE_THRESH) ? p : -1.0f;
        unsigned u   = __float_as_uint(s);
        unsigned key = (u & 0x80000000u) ? ~u : (u | 0x80000000u);
        cr[c] = key;
    };
    emit(lane, e0);                               // classes 0..31
    if (lane + 32 < K_CLS) emit(lane + 32, e1);   // 32..63
    if (lane < 16)         emit(lane + 64, e2);   // 64..79 (80 = bg dropped)

    // box decode (Box2BoxTransform.apply_deltas + clip)
    const float* pr = proposals + (size_t)row * 4;
    float x1 = pr[0], y1 = pr[1];
    float pw = pr[2] - x1, ph = pr[3] - y1;
    float pcx = x1 + 0.5f * pw, pcy = y1 + 0.5f * ph;
    const float* dl = deltas + (size_t)row * DELTAS_LD;
    float* bx = boxes + (size_t)row * DELTAS_LD;
    for (int c = lane; c < K_CLS; c += 32) {
        float dx = dl[c * 4 + 0] * 0.1f;
        float dy = dl[c * 4 + 1] * 0.1f;
        float dw = fminf(dl[c * 4 + 2] * 0.2f, SCALE_CLAMP);
        float dh = fminf(dl[c * 4 + 3] * 0.2f, SCALE_CLAMP);
        float cx = dx * pw + pcx;
        float cy = dy * ph + pcy;
        float w  = __expf(dw) * pw;
        float h  = __expf(dh) * ph;
        bx[c * 4 + 0] = fminf(fmaxf(cx - 0.5f * w, 0.f), IMG_Wc);
        bx[c * 4 + 1] = fminf(fmaxf(cy - 0.5f * h, 0.f), IMG_Hc);
        bx[c * 4 + 2] = fminf(fmaxf(cx + 0.5f * w, 0.f), IMG_Wc);
        bx[c * 4 + 3] = fminf(fmaxf(cy + 0.5f * h, 0.f), IMG_Hc);
    }
}

// ---------------------------------------------------------------------------
// Kernel 3: exact top-1024 per image via 4-pass MSB radix select + compaction.
// Emits (key<<32 | ~idx) pairs (unsorted).
// ---------------------------------------------------------------------------
__global__ __launch_bounds__(256) void radix_select(
    const unsigned int* __restrict__ cand, unsigned long long* __restrict__ selected)
{
    __shared__ unsigned hist[256];
    __shared__ unsigned s_pref;
    __shared__ int s_rem;
    __shared__ unsigned s_cnt, s_eq;
    const int b   = blockIdx.x;
    const int tid = threadIdx.x;
    const unsigned* keys = cand + (size_t)b * CAND_PER_IMG;

    if (tid == 0) { s_pref = 0u; s_rem = M_PRE; }
    __syncthreads();

    for (int p = 3; p >= 0; --p) {
        hist[tid] = 0;
        __syncthreads();
        unsigned pref = s_pref;
        for (int i = tid; i < CAND_PER_IMG; i += 256) {
            unsigned u = keys[i];
            bool m = (p == 3) || ((u >> ((p + 1) * 8)) == pref);
            if (m) atomicAdd(&hist[(u >> (p * 8)) & 255], 1u);
        }
        __syncthreads();
        if (tid == 0) {
            unsigned cum = 0; int v = 255;
            for (; v > 0; --v) { if (cum + hist[v] >= (unsigned)s_rem) break; cum += hist[v]; }
            s_rem  -= (int)cum;
            s_pref  = (s_pref << 8) | (unsigned)v;
        }
        __syncthreads();
    }
    const unsigned kth  = s_pref;
    const int      need = s_rem;          // equal-key quota
    if (tid == 0) { s_cnt = 0; s_eq = 0; }
    __syncthreads();

    unsigned long long* sel = selected + (size_t)b * M_PRE;
    for (int i = tid; i < CAND_PER_IMG; i += 256) {
        unsigned u = keys[i];
        if (u > kth) {
            unsigned pos = atomicAdd(&s_cnt, 1u);
            sel[pos] = ((unsigned long long)u << 32) | (unsigned)(~(unsigned)i);
        } else if (u == kth) {
            unsigned e = atomicAdd(&s_eq, 1u);
            if ((int)e < need) {
                unsigned pos = atomicAdd(&s_cnt, 1u);
                sel[pos] = ((unsigned long long)u << 32) | (unsigned)(~(unsigned)i);
            }
        }
    }
}

// ---------------------------------------------------------------------------
// Kernel 4: per image — bitonic sort 1024 pairs desc (tie: lowest idx first),
// greedy NMS on class-offset boxes in LDS, top-100 compaction, write outputs.
// ---------------------------------------------------------------------------
__global__ __launch_bounds__(256) void sort_nms_out(
    const unsigned long long* __restrict__ selected,
    const float* __restrict__ boxes,
    float* __restrict__ out_boxes, float* __restrict__ out_scores,
    float* __restrict__ out_cls)
{
    __shared__ unsigned long long arr[M_PRE];   // 8 KB
    __shared__ float obx[M_PRE * 4];            // 16 KB (class-offset boxes)
    __shared__ float area_s[M_PRE];             // 4 KB
    __shared__ float val_s[M_PRE];              // 4 KB
    __shared__ int   idx_s[M_PRE];              // 4 KB
    __shared__ int   keep_s[M_PRE];             // 4 KB
    __shared__ int   slot[TOPK];
    const int b   = blockIdx.x;
    const int tid = threadIdx.x;

    for (int i = tid; i < M_PRE; i += 256) arr[i] = selected[(size_t)b * M_PRE + i];

    // bitonic sort, descending by (key, ~idx)
    for (int size = 2; size <= M_PRE; size <<= 1) {
        for (int stride = size >> 1; stride > 0; stride >>= 1) {
            __syncthreads();
            for (int i = tid; i < M_PRE; i += 256) {
                int j = i ^ stride;
                if (j > i) {
                    unsigned long long a = arr[i], c = arr[j];
                    bool descSeg = ((i & size) == 0);
                    bool sw = descSeg ? (a < c) : (a > c);
                    if (sw) { arr[i] = c; arr[j] = a; }
                }
            }
        }
    }
    __syncthreads();

    for (int t = tid; t < M_PRE; t += 256) {
        unsigned long long e = arr[t];
        unsigned key = (unsigned)(e >> 32);
        int idx = (int)(~(unsigned)(e & 0xFFFFFFFFu));
        unsigned u = (key & 0x80000000u) ? (key ^ 0x80000000u) : ~key;
        float val = __uint_as_float(u);
        int cls = idx % K_CLS;
        int n   = idx / K_CLS;
        const float* bp = boxes + (size_t)(b * N_PROP + n) * DELTAS_LD + cls * 4;
        float off = (float)cls * (IMG_Wc + IMG_Hc);   // batched-NMS class offset
        float bx0 = bp[0] + off, by0 = bp[1] + off;
        float bx1 = bp[2] + off, by1 = bp[3] + off;
        obx[t * 4 + 0] = bx0; obx[t * 4 + 1] = by0;
        obx[t * 4 + 2] = bx1; obx[t * 4 + 3] = by1;
        area_s[t] = (bx1 - bx0) * (by1 - by0);
        val_s[t]  = val;
        idx_s[t]  = idx;
        keep_s[t] = (val > SCORE_THRESH) ? 1 : 0;
    }
    __syncthreads();

    // greedy NMS: serial over i, parallel over j
    for (int i = 0; i < M_PRE; ++i) {
        if (keep_s[i]) {
            float ix0 = obx[i * 4], iy0 = obx[i * 4 + 1];
            float ix1 = obx[i * 4 + 2], iy1 = obx[i * 4 + 3];
            float ia = area_s[i];
            for (int j = i + 1 + tid; j < M_PRE; j += 256) {
                if (keep_s[j]) {
                    float xx0 = fmaxf(ix0, obx[j * 4]);
                    float yy0 = fmaxf(iy0, obx[j * 4 + 1]);
                    float xx1 = fminf(ix1, obx[j * 4 + 2]);
                    float yy1 = fminf(iy1, obx[j * 4 + 3]);
                    float w = fmaxf(xx1 - xx0, 0.f), h = fmaxf(yy1 - yy0, 0.f);
                    float inter = w * h;
                    float iou = inter / (ia + area_s[j] - inter + 1e-9f);
                    if (iou > NMS_THRESH) keep_s[j] = 0;
                }
            }
        }
        __syncthreads();
    }

    // top-100: kept (already score-sorted) then non-kept (-1 filler, idx order)
    if (tid == 0) {
        int outn = 0;
        for (int t = 0; t < M_PRE && outn < TOPK; ++t) if (keep_s[t])  slot[outn++] = t;
        for (int t = 0; t < M_PRE && outn < TOPK; ++t) if (!keep_s[t]) slot[outn++] = t;
    }
    __syncthreads();
    if (tid < TOPK) {
        int s   = slot[tid];
        int idx = idx_s[s];
        int cls = idx % K_CLS;
        int n   = idx / K_CLS;
        const float* bp = boxes + (size_t)(b * N_PROP + n) * DELTAS_LD + cls * 4;
        float* ob = out_boxes + ((size_t)b * TOPK + tid) * 4;
        ob[0] = bp[0]; ob[1] = bp[1]; ob[2] = bp[2]; ob[3] = bp[3];
        out_scores[b * TOPK + tid] = keep_s[s] ? val_s[s] : -1.0f;
        out_cls[b * TOPK + tid]    = (float)cls;
    }
}

// ---------------------------------------------------------------------------
extern "C" void kernel_launch(void* const* d_in, const int* in_sizes, int n_in,
                              void* d_out, int out_size, void* d_ws, size_t ws_size,
                              hipStream_t stream) {
    const float* x         = (const float*)d_in[0];
    const float* cls_w     = (const float*)d_in[1];
    const float* cls_b     = (const float*)d_in[2];
    const float* bbox_w    = (const float*)d_in[3];
    const float* bbox_b    = (const float*)d_in[4];
    const float* proposals = (const float*)d_in[5];

    char* ws = (char*)d_ws;
    float*              logits   = (float*)(ws);                         // 16384*96 f32
    float*              deltas   = (float*)(ws + 6291456);               // 16384*320 f32
    float*              boxes    = (float*)(ws + 27262976);              // 16384*320 f32
    unsigned int*       cand     = (unsigned int*)(ws + 48234496);       // 16*81920 u32
    unsigned long long* selected = (unsigned long long*)(ws + 53477376); // 16*1024 u64

    float* out_boxes  = (float*)d_out;          // [16,100,4]
    float* out_scores = out_boxes + B_IMG * TOPK * 4;
    float* out_cls    = out_scores + B_IMG * TOPK;

    gemm_head<<<dim3(M_ROWS / BM, LOGITS_LD / BN), 256, 0, stream>>>(
        x, cls_w, cls_b, logits, K_CLS + 1, LOGITS_LD);
    gemm_head<<<dim3(M_ROWS / BM, DELTAS_LD / BN), 256, 0, stream>>>(
        x, bbox_w, bbox_b, deltas, 4 * K_CLS, DELTAS_LD);
    softmax_decode<<<M_ROWS / 8, 256, 0, stream>>>(logits, deltas, proposals, boxes, cand);
    radix_select<<<B_IMG, 256, 0, stream>>>(cand, selected);
    sort_nms_out<<<B_IMG, 256, 0, stream>>>(selected, boxes, out_boxes, out_scores, out_cls);
}